// AdaLoRA_48576080118359
// MI455X (gfx1250) — compile-verified
//
#include <hip/hip_runtime.h>
#include <stdint.h>

typedef __attribute__((ext_vector_type(2))) float v2f;
typedef __attribute__((ext_vector_type(4))) float v4f;
typedef __attribute__((ext_vector_type(8))) float v8f;
typedef __attribute__((ext_vector_type(4))) int   v4i;

#define IN_F   4096
#define OUT_F  4096
#define RANK   12
#define SCALING (16.0f / 12.0f)

#define BM   128
#define BN   128
#define BK   32
#define KPAD 36   // 36*4=144B rows: 16B-aligned, conflict-free 16-lane column reads

#if defined(__AMDGCN__) && \
    __has_builtin(__builtin_amdgcn_global_load_async_to_lds_b128) && \
    __has_builtin(__builtin_amdgcn_s_wait_asynccnt)
#define USE_ASYNC_LDS 1
#else
#define USE_ASYNC_LDS 0
#endif

typedef __attribute__((address_space(1))) v4i gbl_v4i;  // global (b128 payload)
typedef __attribute__((address_space(3))) v4i lds_v4i;  // LDS    (b128 payload)

__device__ __forceinline__ void async_copy_b128(const float* gsrc, float* ldst) {
#if USE_ASYNC_LDS
    // flat-LDS addr[31:0] is the LDS offset,
    // so truncating the generic pointer through uintptr_t is the AS3 address.
    __builtin_amdgcn_global_load_async_to_lds_b128(
        (gbl_v4i*)(uintptr_t)gsrc,
        (lds_v4i*)(uint32_t)(uintptr_t)ldst,
        0, 0);
#else
    *(v4f*)ldst = *(const v4f*)gsrc;
#endif
}

__device__ __forceinline__ void async_wait_all() {
#if USE_ASYNC_LDS
    __builtin_amdgcn_s_wait_asynccnt(0);
#endif
}

// ---------------------------------------------------------------------------
// Kernel A: T[m, r] = SCALING * sigma[r] * sum_k x[m,k] * Q[k,r]
// One wave per row m; coalesced x row stream; wave32 shfl_xor reduction.
// ---------------------------------------------------------------------------
__global__ __launch_bounds__(256)
void adalora_xq_kernel(const float* __restrict__ x,
                       const float* __restrict__ Q,
                       const float* __restrict__ sigma,
                       float* __restrict__ T, int M) {
    const int lane = threadIdx.x & 31;
    const int wave = threadIdx.x >> 5;
    const int m = blockIdx.x * 8 + wave;
    if (m >= M) return;

    float acc[RANK];
#pragma unroll
    for (int r = 0; r < RANK; ++r) acc[r] = 0.0f;

    const float* __restrict__ xrow = x + (size_t)m * IN_F;
    for (int k = lane; k < IN_F; k += 32) {
        const float xv = xrow[k];
        const float* __restrict__ q = Q + (size_t)k * RANK;
#pragma unroll
        for (int r = 0; r < RANK; ++r) acc[r] = fmaf(xv, q[r], acc[r]);
    }

#pragma unroll
    for (int r = 0; r < RANK; ++r) {
        float v = acc[r];
#pragma unroll
        for (int off = 16; off > 0; off >>= 1) v += __shfl_xor(v, off, 32);
        acc[r] = v;
    }

    if (lane == 0) {
#pragma unroll
        for (int r = 0; r < RANK; ++r)
            T[(size_t)m * RANK + r] = SCALING * sigma[r] * acc[r];
    }
}

// ---------------------------------------------------------------------------
// Kernel B: out = x @ W^T (+ T @ P^T as one extra zero-padded K-iteration)
// 128x128 block tile, BK=32, 8 waves = 4(M) x 2(N); each wave: 2x4 16x16 tiles.
// Double-buffered LDS filled by GLOBAL_LOAD_ASYNC_TO_LDS_B128 (ASYNCcnt),
// one s_wait_asynccnt+barrier per K-tile.
// ---------------------------------------------------------------------------
__global__ __launch_bounds__(256)
void adalora_gemm_kernel(const float* __restrict__ x,
                         const float* __restrict__ W,
                         const float* __restrict__ T,
                         const float* __restrict__ P,
                         float* __restrict__ out, int M) {
    __shared__ float xs[2][BM * KPAD];
    __shared__ float ws[2][BN * KPAD];

    const int tid  = threadIdx.x;
    const int lane = tid & 31;
    const int wave = tid >> 5;
    const int wm   = wave & 3;    // 32-row subtile
    const int wn   = wave >> 2;   // 64-col subtile
    const int m0   = blockIdx.y * BM;
    const int n0   = blockIdx.x * BN;
    const int half = lane >> 4;   // 0: lanes 0-15, 1: lanes 16-31
    const int l16  = lane & 15;

    const v8f vzero = {0.f, 0.f, 0.f, 0.f, 0.f, 0.f, 0.f, 0.f};
    v8f c[2][4];
#pragma unroll
    for (int mt = 0; mt < 2; ++mt)
#pragma unroll
        for (int nt = 0; nt < 4; ++nt) c[mt][nt] = vzero;

    const int r32 = tid >> 3;         // 0..31 rows per pass
    const int c4  = (tid & 7) * 4;    // float4 column within 32-wide K tile

    // Stage one 128x32 x-tile and 128x32 W-tile into LDS buffer `buf`.
    auto issue_tile = [&](int k0, int buf) {
#pragma unroll
        for (int p = 0; p < 4; ++p) {
            const int row = p * 32 + r32;
            async_copy_b128(x + (size_t)(m0 + row) * IN_F + k0 + c4,
                            &xs[buf][row * KPAD + c4]);
            async_copy_b128(W + (size_t)(n0 + row) * IN_F + k0 + c4,
                            &ws[buf][row * KPAD + c4]);
        }
    };

    // One WMMA step at LDS K-offset k (covers K=k..k+3); ISA 16x4 f32 layout:
    // lanes 0-15 read (K=k,k+1), lanes 16-31 read (K=k+2,k+3).
    auto mma_step = [&](const float* xb, const float* wb, int k) {
        v2f a[2], b[4];
#pragma unroll
        for (int mt = 0; mt < 2; ++mt) {
            const int row = wm * 32 + mt * 16 + l16;
            a[mt] = *(const v2f*)&xb[row * KPAD + k + half * 2];
        }
#pragma unroll
        for (int nt = 0; nt < 4; ++nt) {
            const int col = wn * 64 + nt * 16 + l16;
            b[nt] = *(const v2f*)&wb[col * KPAD + k + half * 2];
        }
#pragma unroll
        for (int mt = 0; mt < 2; ++mt)
#pragma unroll
            for (int nt = 0; nt < 4; ++nt)
                c[mt][nt] = __builtin_amdgcn_wmma_f32_16x16x4_f32(
                    false, a[mt], false, b[nt], (short)0, c[mt][nt],
                    false, false);
    };

    // Prologue: fill buffer 0.
    issue_tile(0, 0);
    async_wait_all();
    __syncthreads();

    for (int k0 = 0; k0 < IN_F; k0 += BK) {
        const int buf = (k0 >> 5) & 1;
        if (k0 + BK < IN_F) issue_tile(k0 + BK, buf ^ 1);  // prefetch next tile
#pragma unroll
        for (int k = 0; k < BK; k += 4)
            mma_step(&xs[buf][0], &ws[buf][0], k);
        async_wait_all();   // own async loads for next buffer complete
        __syncthreads();    // all waves done: next buffer valid, this one free
    }

    // LoRA epilogue as one extra zero-padded K-iteration: A=T tile, B=P tile.
    for (int idx = tid; idx < BM * BK; idx += 256) {
        const int row = idx >> 5;
        const int cc  = idx & 31;
        xs[0][row * KPAD + cc] = (cc < RANK) ? T[(size_t)(m0 + row) * RANK + cc] : 0.0f;
        ws[0][row * KPAD + cc] = (cc < RANK) ? P[(size_t)(n0 + row) * RANK + cc] : 0.0f;
    }
    __syncthreads();
    mma_step(&xs[0][0], &ws[0][0], 0);
    mma_step(&xs[0][0], &ws[0][0], 4);
    mma_step(&xs[0][0], &ws[0][0], 8);   // rank 0..11; padded cols contribute zero

    // Store: C/D layout -> VGPR j holds rows M=j (lanes 0-15) and M=j+8 (16-31).
#pragma unroll
    for (int mt = 0; mt < 2; ++mt) {
#pragma unroll
        for (int nt = 0; nt < 4; ++nt) {
            const int n     = n0 + wn * 64 + nt * 16 + l16;
            const int mbase = m0 + wm * 32 + mt * 16 + half * 8;
#pragma unroll
            for (int j = 0; j < 8; ++j)
                out[(size_t)(mbase + j) * OUT_F + n] = c[mt][nt][j];
        }
    }
}

// ---------------------------------------------------------------------------
extern "C" void kernel_launch(void* const* d_in, const int* in_sizes, int n_in,
                              void* d_out, int out_size, void* d_ws, size_t ws_size,
                              hipStream_t stream) {
    const float* x     = (const float*)d_in[0];   // [4,4096,4096]
    const float* W     = (const float*)d_in[1];   // [4096,4096]
    const float* P     = (const float*)d_in[2];   // [4096,12]
    const float* sigma = (const float*)d_in[3];   // [12]
    const float* Q     = (const float*)d_in[4];   // [4096,12]
    float* out = (float*)d_out;
    float* T   = (float*)d_ws;                    // [M,12] scratch (786 KB)

    const int M = in_sizes[0] / IN_F;             // 16384

    adalora_xq_kernel<<<dim3((M + 7) / 8), dim3(256), 0, stream>>>(x, Q, sigma, T, M);
    adalora_gemm_kernel<<<dim3(OUT_F / BN, M / BM), dim3(256), 0, stream>>>(
        x, W, T, P, out, M);
}